// NormalLoss_87093346828430
// MI455X (gfx1250) — compile-verified
//
#include <hip/hip_runtime.h>

typedef __attribute__((ext_vector_type(2))) float v2f;
typedef __attribute__((ext_vector_type(8))) float v8f;

#define WAVES_PER_BLOCK 4

// Kernel 1: one wave per 16-row template tile. Sweep scan points in 16-column
// tiles with V_WMMA_F32_16X16X4_F32 (A = 16x4 template coords, B = 4x16 scan
// coords, D = 16x16 dot products). Track per-lane running argmin of
// (s2[n] - 2*dot) — t2[m] is a row constant and drops out of the argmin —
// then do a 4-step shuffle argmin across each 16-lane half at the end.
__global__ __launch_bounds__(32 * WAVES_PER_BLOCK)
void nn_argmin_wmma(const float* __restrict__ tmpl,   // [M,3]
                    const float* __restrict__ scan,   // [N,3]
                    int* __restrict__ idxOut,         // [M]
                    int M, int N)
{
    const int lane = threadIdx.x & 31;
    const int wave = threadIdx.x >> 5;
    const int tileId = blockIdx.x * WAVES_PER_BLOCK + wave;
    const int numTiles = (M + 15) >> 4;
    if (tileId >= numTiles) return;          // wave-uniform: EXEC stays all-1s

    const int mBase = tileId << 4;
    const int col   = lane & 15;             // matrix row (A) / column (B,D)
    const int half  = lane >> 4;

    // A fragment, 16x4 f32: lanes 0-15 hold K=0,1; lanes 16-31 hold K=2,3.
    int mA = mBase + col; if (mA > M - 1) mA = M - 1;
    const float tx = tmpl[3 * mA + 0];
    const float ty = tmpl[3 * mA + 1];
    const float tz = tmpl[3 * mA + 2];
    v2f a;
    a.x = half ? tz : tx;
    a.y = half ? 0.0f : ty;

    float bestV[8];
    int   bestI[8];
#pragma unroll
    for (int r = 0; r < 8; ++r) { bestV[r] = 3.402823466e38f; bestI[r] = 0; }

    const v8f czero = {0.f, 0.f, 0.f, 0.f, 0.f, 0.f, 0.f, 0.f};

    // ---- main loop: full 16-column tiles, fully branchless body ----
    const int nFull = N >> 4;
    const float* sp = scan + 3 * col;        // this lane's column, tile 0
#pragma unroll 2
    for (int t = 0; t < nFull; ++t) {
        const float sx = sp[0];
        const float sy = sp[1];
        const float sz = sp[2];
        // B fragment, 4x16 f32: lanes 0-15 carry K rows 0/1, lanes 16-31 rows 2/3.
        v2f b;
        b.x = half ? sz : sx;
        b.y = half ? 0.0f : sy;
        const float s2 = fmaf(sx, sx, fmaf(sy, sy, sz * sz));

        const v8f d = __builtin_amdgcn_wmma_f32_16x16x4_f32(
            false, a, false, b, (short)0, czero, false, false);

        const int n = (t << 4) + col;
#pragma unroll
        for (int r = 0; r < 8; ++r) {
            const float key = fmaf(-2.0f, d[r], s2);   // s2[n] - 2*dot(m,n)
            const bool better = key < bestV[r];        // strict: first index wins ties
            bestV[r] = better ? key : bestV[r];
            bestI[r] = better ? n : bestI[r];
        }
        sp += 48;                                       // 16 points * 3 floats
    }

    // ---- tail tile (N % 16 != 0); uniform branch, EXEC stays all-1s ----
    if (N & 15) {
        const int n = (nFull << 4) + col;
        const int nc = (n < N) ? n : (N - 1);
        const float sx = scan[3 * nc + 0];
        const float sy = scan[3 * nc + 1];
        const float sz = scan[3 * nc + 2];
        v2f b;
        b.x = half ? sz : sx;
        b.y = half ? 0.0f : sy;
        const float s2 = fmaf(sx, sx, fmaf(sy, sy, sz * sz));

        const v8f d = __builtin_amdgcn_wmma_f32_16x16x4_f32(
            false, a, false, b, (short)0, czero, false, false);

        const bool inR = (n < N);
#pragma unroll
        for (int r = 0; r < 8; ++r) {
            float key = fmaf(-2.0f, d[r], s2);
            key = inR ? key : 3.402823466e38f;
            const bool better = key < bestV[r];
            bestV[r] = better ? key : bestV[r];
            bestI[r] = better ? n : bestI[r];
        }
    }

    // ---- cross-lane argmin within each 16-lane half (xor 1..8 stays in-half).
    // Lower index wins ties, matching jnp.argmin first-occurrence semantics.
#pragma unroll
    for (int r = 0; r < 8; ++r) {
        float v = bestV[r];
        int   i = bestI[r];
#pragma unroll
        for (int off = 1; off < 16; off <<= 1) {
            const float v2 = __shfl_xor(v, off);
            const int   i2 = __shfl_xor(i, off);
            const bool take = (v2 < v) | ((v2 == v) & (i2 < i));
            v = take ? v2 : v;
            i = take ? i2 : i;
        }
        if (col == 0) {
            const int m = mBase + r + (half << 3);   // D layout: VGPR r -> row r + 8*half
            if (m < M) idxOut[m] = i;
        }
    }
}

// Kernel 2: recompute exact NN distance (as the reference does), gate by
// normal agreement (angle < 30deg  <=>  clipped dot > cos(30deg)), and do a
// single-block deterministic tree reduction (no float atomics -> bitwise-
// stable output across graph replays).
__global__ __launch_bounds__(256)
void gate_reduce(const float* __restrict__ tmpl,
                 const float* __restrict__ scan,
                 const float* __restrict__ snrm,
                 const float* __restrict__ tnrm,
                 const int* __restrict__ idx,
                 float* __restrict__ out, int M)
{
    __shared__ float red[256];
    float acc = 0.0f;
    for (int m = threadIdx.x; m < M; m += 256) {
        const int j = idx[m];
        const float dx = tmpl[3 * m + 0] - scan[3 * j + 0];
        const float dy = tmpl[3 * m + 1] - scan[3 * j + 1];
        const float dz = tmpl[3 * m + 2] - scan[3 * j + 2];
        const float dist = fmaf(dx, dx, fmaf(dy, dy, dz * dz));
        float dot = fmaf(snrm[3 * j + 0], tnrm[3 * m + 0],
                    fmaf(snrm[3 * j + 1], tnrm[3 * m + 1],
                         snrm[3 * j + 2] * tnrm[3 * m + 2]));
        dot = fminf(1.0f, fmaxf(-1.0f, dot));
        const float contrib = (dot > 0.86602540378443864676f) ? dist : 0.0f;
        acc += contrib;
    }
    red[threadIdx.x] = acc;
    __syncthreads();
    for (int s = 128; s > 0; s >>= 1) {
        if (threadIdx.x < s) red[threadIdx.x] += red[threadIdx.x + s];
        __syncthreads();
    }
    if (threadIdx.x == 0) out[0] = red[0];
}

extern "C" void kernel_launch(void* const* d_in, const int* in_sizes, int n_in,
                              void* d_out, int out_size, void* d_ws, size_t ws_size,
                              hipStream_t stream) {
    const float* scan = (const float*)d_in[0];   // scan_vertices     [N,3]
    const float* tmpl = (const float*)d_in[1];   // template_vertices [M,3]
    const float* snrm = (const float*)d_in[2];   // scan_normals      [N,3]
    const float* tnrm = (const float*)d_in[3];   // template_normals  [M,3]
    const int N = in_sizes[0] / 3;
    const int M = in_sizes[1] / 3;

    int* idx = (int*)d_ws;                       // M ints of scratch

    const int numTiles = (M + 15) / 16;
    const int blocks = (numTiles + WAVES_PER_BLOCK - 1) / WAVES_PER_BLOCK;

    hipLaunchKernelGGL(nn_argmin_wmma, dim3(blocks), dim3(32 * WAVES_PER_BLOCK),
                       0, stream, tmpl, scan, idx, M, N);
    hipLaunchKernelGGL(gate_reduce, dim3(1), dim3(256), 0, stream,
                       tmpl, scan, snrm, tnrm, idx, (float*)d_out, M);
}